// BiochemicalDiffusion_30923764531786
// MI455X (gfx1250) — compile-verified
//
#include <hip/hip_runtime.h>

// dx/dt = 1.0 - 0.1*x - 0.01 * rowsum(x * (A@x))
// Dominant cost: C = A @ x GEMM, M=K=10000, N=64, fp32.
// HBM floor: stream A once = 400 MB @ 23.3 TB/s ~ 17.2 us. Needs ~745 TFLOPS
// fp32 to stay memory-bound -> V_WMMA_F32_16X16X4_F32 (full fp32 accuracy).
// B-operand (x) is pre-packed into fragment order in d_ws so each B fragment
// is one coalesced global_load_b64; wave id goes through readfirstlane so the
// K loop is a uniform scalar loop (EXEC stays all-1s for WMMA).

#define NROWS 10000
#define DIM   64

typedef float v2f __attribute__((ext_vector_type(2)));
typedef float v8f __attribute__((ext_vector_type(8)));

// ---------------------------------------------------------------------------
// Pack x[10000][64] into B-fragment order:
//   xpack[kstep*256 + khalf*128 + n*2 + p] = x[4*kstep + 2*khalf + p][n]
// so a lane-addressed b64 load yields (x[k'][n], x[k'+1][n]) directly.
// ---------------------------------------------------------------------------
__global__ __launch_bounds__(256) void
pack_x_kernel(const float* __restrict__ x, float* __restrict__ xpack) {
    const int idx   = blockIdx.x * 256 + threadIdx.x;  // 0 .. 640000-1
    const int kstep = idx >> 8;
    const int rem   = idx & 255;
    const int khalf = rem >> 7;
    const int n     = (rem >> 1) & 63;
    const int p     = rem & 1;
    const int krow  = kstep * 4 + khalf * 2 + p;
    xpack[idx] = x[(size_t)krow * DIM + n];
}

// ---------------------------------------------------------------------------
// Fused GEMM + epilogue. One workgroup per 16-row M-tile; 8 waves split K
// (stride 32), reduce partials via ds_add_f32.
// ---------------------------------------------------------------------------
template <bool PACKED>
__global__ __launch_bounds__(256) void
biochem_fused_wmma(const float* __restrict__ x,
                   const float* __restrict__ A,
                   const float* __restrict__ xpack,
                   float* __restrict__ out) {
    __shared__ float Csh[16][DIM];   // accumulated C = (A@x) tile, 4 KB
    __shared__ float srow[16];       // per-row interaction sums

    const int tid   = threadIdx.x;
    const int lane  = tid & 31;
    const int wave  = __builtin_amdgcn_readfirstlane(tid >> 5);  // uniform SGPR
    const int lhalf = lane >> 4;     // 0/1: selects K pair within fragment
    const int l16   = lane & 15;
    const int m0    = blockIdx.x * 16;

    for (int i = tid; i < 16 * DIM; i += 256) ((float*)Csh)[i] = 0.0f;
    __syncthreads();

    v8f acc0 = {}; v8f acc1 = {}; v8f acc2 = {}; v8f acc3 = {};

    // A fragment (16x4 fp32): lane 0-15 -> A[m0+l][k..k+1],
    //                         lane 16-31 -> A[m0+l-16][k+2..k+3]
    const float* Aptr = A + (size_t)(m0 + l16) * NROWS + 2 * lhalf + 4 * wave;
    // Packed B: v2f element offset = kstep*128 + lhalf*64 + (n0 + l16)
    const v2f* bp = (const v2f*)xpack + (size_t)wave * 128 + lhalf * 64 + l16;

    #pragma unroll 4
    for (int k = 4 * wave; k < NROWS; k += 32) {   // scalar uniform loop
        v2f a = *(const v2f*)Aptr;                  // 8B-aligned
        Aptr += 32;

        v2f b0, b1, b2, b3;
        if constexpr (PACKED) {
            b0 = bp[0];  b1 = bp[16];  b2 = bp[32];  b3 = bp[48];
            bp += 1024;                             // 8 ksteps * 128 v2f
        } else {
            const int    kr = k + 2 * lhalf;
            const float* xr = x + (size_t)kr * DIM + l16;
            b0.x = xr[0];   b0.y = xr[DIM + 0];
            b1.x = xr[16];  b1.y = xr[DIM + 16];
            b2.x = xr[32];  b2.y = xr[DIM + 32];
            b3.x = xr[48];  b3.y = xr[DIM + 48];
        }

        acc0 = __builtin_amdgcn_wmma_f32_16x16x4_f32(false, a, false, b0,
                                                     (short)0, acc0, false, false);
        acc1 = __builtin_amdgcn_wmma_f32_16x16x4_f32(false, a, false, b1,
                                                     (short)0, acc1, false, false);
        acc2 = __builtin_amdgcn_wmma_f32_16x16x4_f32(false, a, false, b2,
                                                     (short)0, acc2, false, false);
        acc3 = __builtin_amdgcn_wmma_f32_16x16x4_f32(false, a, false, b3,
                                                     (short)0, acc3, false, false);
    }

    // Cross-wave reduction. D VGPR r: lanes0-15 = C[r][lane],
    // lanes16-31 = C[8+r][lane-16]  ->  ds_add_f32 into Csh.
    {
        const int mrow = 8 * lhalf;
        #pragma unroll
        for (int r = 0; r < 8; ++r) {
            atomicAdd(&Csh[mrow + r][l16 +  0], acc0[r]);
            atomicAdd(&Csh[mrow + r][l16 + 16], acc1[r]);
            atomicAdd(&Csh[mrow + r][l16 + 32], acc2[r]);
            atomicAdd(&Csh[mrow + r][l16 + 48], acc3[r]);
        }
    }
    __syncthreads();

    // Per-row interaction term s_i = <x_i, (A@x)_i>
    if (tid < 16) {
        const float4* xr = (const float4*)(x + (size_t)(m0 + tid) * DIM);
        const float4* cr = (const float4*)(&Csh[tid][0]);
        float s = 0.0f;
        #pragma unroll
        for (int j = 0; j < DIM / 4; ++j) {
            float4 xv = xr[j];
            float4 cv = cr[j];
            s += xv.x * cv.x + xv.y * cv.y + xv.z * cv.z + xv.w * cv.w;
        }
        srow[tid] = s;
    }
    __syncthreads();

    // Epilogue: out = 1.0 - 0.1*x - 0.01*s_row, one float4 per thread.
    {
        const int r = tid >> 4;          // 0..15
        const int c = (tid & 15) * 4;    // 0..60
        const size_t off = (size_t)(m0 + r) * DIM + c;
        const float4 xv = *(const float4*)(x + off);
        const float  s  = 0.01f * srow[r];
        float4 o;
        o.x = 1.0f - 0.1f * xv.x - s;
        o.y = 1.0f - 0.1f * xv.y - s;
        o.z = 1.0f - 0.1f * xv.z - s;
        o.w = 1.0f - 0.1f * xv.w - s;
        *(float4*)(out + off) = o;
    }
}

extern "C" void kernel_launch(void* const* d_in, const int* in_sizes, int n_in,
                              void* d_out, int out_size, void* d_ws, size_t ws_size,
                              hipStream_t stream) {
    // setup_inputs order: t (unused, 1 elem), x [N,DIM] f32, A [N,N] f32
    const float* x = (const float*)d_in[1];
    const float* A = (const float*)d_in[2];
    float* out = (float*)d_out;

    (void)in_sizes; (void)n_in; (void)out_size;

    const size_t packBytes = (size_t)NROWS * DIM * sizeof(float);  // 2.56 MB
    if (ws_size >= packBytes && d_ws != nullptr) {
        float* xpack = (float*)d_ws;
        pack_x_kernel<<<NROWS / 4, 256, 0, stream>>>(x, xpack);
        biochem_fused_wmma<true><<<NROWS / 16, 256, 0, stream>>>(x, A, xpack, out);
    } else {
        biochem_fused_wmma<false><<<NROWS / 16, 256, 0, stream>>>(x, A, x, out);
    }
}